// LinkPrediction_39393440039448
// MI455X (gfx1250) — compile-verified
//
#include <hip/hip_runtime.h>

#define N_NODES   50000
#define N_REL     10
#define R2        (2*N_REL + 1)
#define N_TRIPLES 500000
#define BATCH     16384
#define HID       64
#define TILES     (N_NODES/16)   // 3125, exact (no tail)

typedef float v2f __attribute__((ext_vector_type(2)));
typedef float v8f __attribute__((ext_vector_type(8)));
typedef int   v4i_ls __attribute__((vector_size(4 * sizeof(int))));  // builtin's pointee type

#if __has_builtin(__builtin_amdgcn_global_load_async_to_lds_b128)
#define HAVE_ASYNC_LDS 1
#else
#define HAVE_ASYNC_LDS 0
#endif

__device__ __forceinline__ void wait_asynccnt0() {
#if __has_builtin(__builtin_amdgcn_s_wait_asynccnt)
  __builtin_amdgcn_s_wait_asynccnt(0);
#else
  asm volatile("s_wait_asynccnt 0x0" ::: "memory");
#endif
}

// ---- init: zero colsum, h = b0 + self-loop w0 row (vals==1), nodes = b1 ----
__global__ void k_init(float* __restrict__ colsum, float* __restrict__ h,
                       float* __restrict__ nodes, const float* __restrict__ w0,
                       const float* __restrict__ b0, const float* __restrict__ b1) {
  int i = blockIdx.x * blockDim.x + threadIdx.x;
  if (i < 2*N_REL*N_NODES) colsum[i] = 0.0f;
  if (i < N_NODES*HID) {
    int c = i & 63;
    nodes[i] = b1[c];
    h[i] = b0[c] + w0[(long)(2*N_REL)*N_NODES*HID + i];  // rel 2r, node i/64
  }
}

// ---- colsum: count edges per (relation, object) column ----
__global__ void k_colsum(const int* __restrict__ tri, float* __restrict__ colsum) {
  int e = blockIdx.x * blockDim.x + threadIdx.x;
  if (e >= N_TRIPLES) return;
  int s = tri[e*3 + 0], p = tri[e*3 + 1], o = tri[e*3 + 2];
  atomicAdd(&colsum[p*N_NODES + o], 1.0f);                 // forward edge
  atomicAdd(&colsum[(p + N_REL)*N_NODES + s], 1.0f);       // inverse edge
}

// ---- layer 0: h[s_all] += vals * w0[col]; 64 threads per triple ----
__global__ void k_layer0(const int* __restrict__ tri, const float* __restrict__ colsum,
                         const float* __restrict__ w0, float* __restrict__ h) {
  long t = (long)blockIdx.x * blockDim.x + threadIdx.x;
  long e = t >> 6;
  int  c = (int)(t & 63);
  if (e >= N_TRIPLES) return;
  int s = tri[e*3 + 0], p = tri[e*3 + 1], o = tri[e*3 + 2];
  int colF = p*N_NODES + o;
  int colI = (p + N_REL)*N_NODES + s;
  float vF = 1.0f / colsum[colF];
  float vI = 1.0f / colsum[colI];
  atomicAdd(&h[s*HID + c], vF * w0[(long)colF*HID + c]);
  atomicAdd(&h[o*HID + c], vI * w0[(long)colI*HID + c]);
}

__global__ void k_zero(float* __restrict__ m) {
  int i = blockIdx.x * blockDim.x + threadIdx.x;
  if (i < N_NODES*HID) m[i] = 0.0f;
}

// ---- layer-1 pre-aggregation for relation p: m[tgt] += vals * h[src] ----
__global__ void k_scatter(const int* __restrict__ tri, const float* __restrict__ colsum,
                          const float* __restrict__ h, float* __restrict__ m, int p) {
  long t = (long)blockIdx.x * blockDim.x + threadIdx.x;
  long e = t >> 6;
  int  c = (int)(t & 63);
  if (e >= N_TRIPLES) return;
  int pt = tri[e*3 + 1];
  int tgt, src, col;
  if (p < N_REL) {                 // forward edges of relation p
    if (pt != p) return;
    int s = tri[e*3 + 0], o = tri[e*3 + 2];
    tgt = s; src = o; col = p*N_NODES + o;
  } else {                         // inverse edges of relation p-r
    if (pt != p - N_REL) return;
    int s = tri[e*3 + 0], o = tri[e*3 + 2];
    tgt = o; src = s; col = p*N_NODES + s;
  }
  float v = 1.0f / colsum[col];
  atomicAdd(&m[tgt*HID + c], v * h[src*HID + c]);
}

// ---- WMMA GEMM: nodes(50000x64) += A(50000x64) @ Bmat(64x64), fp32 ----
// One wave per 16-row tile; 4 accumulator tiles of 16x16; K=64 in 16 steps of 4.
// Bmat is staged into LDS with the gfx1250 async global->LDS path when available.
__global__ __launch_bounds__(256) void k_gemm(const float* __restrict__ A,
                                              const float* __restrict__ Bmat,
                                              float* __restrict__ nodes) {
  __shared__ float Bs[HID*HID];

#if HAVE_ASYNC_LDS
  // 256 threads x 4 iters x 16B = 16 KB, ASYNCcnt-tracked DMA into LDS
  for (int i = threadIdx.x; i < (HID*HID)/4; i += 256) {
    __builtin_amdgcn_global_load_async_to_lds_b128(
        (__attribute__((address_space(1))) v4i_ls*)(Bmat + i*4),
        (__attribute__((address_space(3))) v4i_ls*)(&Bs[i*4]),
        0, 0);
  }
  wait_asynccnt0();
#else
  for (int i = threadIdx.x; i < HID*HID; i += 256) Bs[i] = Bmat[i];
#endif
  __syncthreads();

  int wave = threadIdx.x >> 5;
  int lane = threadIdx.x & 31;
  int tile = blockIdx.x * 8 + wave;
  if (tile >= TILES) return;       // uniform per wave; EXEC all-ones for WMMA

  int half  = lane >> 4;           // 0: lanes 0-15, 1: lanes 16-31
  int ml    = lane & 15;
  int Mbase = tile * 16;

  // C/D layout: VGPR v, lane -> M = Mbase + v + 8*half, N = nt*16 + ml
  v8f acc[4];
#pragma unroll
  for (int nt = 0; nt < 4; ++nt)
#pragma unroll
    for (int v = 0; v < 8; ++v)
      acc[nt][v] = nodes[(Mbase + v + 8*half)*HID + nt*16 + ml];

#pragma unroll
  for (int kk = 0; kk < 16; ++kk) {
    int kbase = kk * 4;
    // A 16x4 layout: lane ml holds K = kbase + 2*half + {0,1}
    const float* arow = A + (long)(Mbase + ml)*HID + kbase + 2*half;
    v2f a; a.x = arow[0]; a.y = arow[1];
#pragma unroll
    for (int nt = 0; nt < 4; ++nt) {
      int bcol = nt*16 + ml;
      // B 4x16 layout: VGPR v holds K-row = 2*half + v
      v2f b;
      b.x = Bs[(kbase + 2*half + 0)*HID + bcol];
      b.y = Bs[(kbase + 2*half + 1)*HID + bcol];
      acc[nt] = __builtin_amdgcn_wmma_f32_16x16x4_f32(
          false, a, false, b, (short)0, acc[nt], false, false);
    }
  }

#pragma unroll
  for (int nt = 0; nt < 4; ++nt)
#pragma unroll
    for (int v = 0; v < 8; ++v)
      nodes[(Mbase + v + 8*half)*HID + nt*16 + ml] = acc[nt][v];
}

// ---- scores: one wave per batch element, shuffle reduction ----
__global__ void k_scores(const int* __restrict__ batch, const float* __restrict__ nodes,
                         const float* __restrict__ rel, float* __restrict__ out) {
  int wid  = (blockIdx.x * blockDim.x + threadIdx.x) >> 5;
  int lane = threadIdx.x & 31;
  if (wid >= BATCH) return;
  int si = batch[wid*3 + 0], pi = batch[wid*3 + 1], oi = batch[wid*3 + 2];
  float v = 0.0f;
#pragma unroll
  for (int k = 0; k < 2; ++k) {
    int c = lane + 32*k;
    v += nodes[si*HID + c] * rel[pi*HID + c] * nodes[oi*HID + c];
  }
  for (int off = 16; off > 0; off >>= 1) v += __shfl_xor(v, off, 32);
  if (lane == 0) out[wid] = v;
}

extern "C" void kernel_launch(void* const* d_in, const int* in_sizes, int n_in,
                              void* d_out, int out_size, void* d_ws, size_t ws_size,
                              hipStream_t stream) {
  const int*   batch = (const int*)d_in[0];
  const int*   tri   = (const int*)d_in[1];
  const float* w0    = (const float*)d_in[2];
  const float* b0    = (const float*)d_in[3];
  const float* w1    = (const float*)d_in[4];
  const float* b1    = (const float*)d_in[5];
  const float* rel   = (const float*)d_in[6];
  float* out = (float*)d_out;

  // workspace layout (floats): colsum[1.0M] | m[3.2M] | h[3.2M] | nodes[3.2M] = ~42.4 MB
  float* colsum = (float*)d_ws;
  float* m      = colsum + (size_t)2*N_REL*N_NODES;
  float* h      = m + (size_t)N_NODES*HID;
  float* nodes  = h + (size_t)N_NODES*HID;

  k_init  <<<(N_NODES*HID + 255)/256, 256, 0, stream>>>(colsum, h, nodes, w0, b0, b1);
  k_colsum<<<(N_TRIPLES + 255)/256,   256, 0, stream>>>(tri, colsum);
  k_layer0<<<(N_TRIPLES*64)/256,      256, 0, stream>>>(tri, colsum, w0, h);

  for (int p = 0; p < 2*N_REL; ++p) {
    k_zero   <<<(N_NODES*HID + 255)/256, 256, 0, stream>>>(m);
    k_scatter<<<(N_TRIPLES*64)/256,      256, 0, stream>>>(tri, colsum, h, m, p);
    k_gemm   <<<(TILES + 7)/8,           256, 0, stream>>>(m, w1 + (size_t)p*HID*HID, nodes);
  }
  // self-loop relation 2r: vals == 1, m_p == h
  k_gemm<<<(TILES + 7)/8, 256, 0, stream>>>(h, w1 + (size_t)(2*N_REL)*HID*HID, nodes);

  k_scores<<<(BATCH*32)/256, 256, 0, stream>>>(batch, nodes, rel, out);
}